// feature_extractor_74904229642789
// MI455X (gfx1250) — compile-verified
//
#include <hip/hip_runtime.h>
#include <math.h>

#define IMG 512
#define NPIX (IMG * IMG)
#define NCH 48
#define RMAX 134
#define SLABW2 (2 * RMAX + 132)  // 400: 128-wide block + 2r halo + K padding
#define GP_STRIDE 352            // per-octave padded weights (16 zero guard + 2r+1 + zeros)
#define HARRIS_K 0.04f
#define THRESH 0.01f

typedef __attribute__((ext_vector_type(2))) float v2f;
typedef __attribute__((ext_vector_type(8))) float v8f;

__device__ __forceinline__ int refl(int p) {          // reflect-101 on [0,512)
  p = (p < 0) ? -p : p;
  return (p >= IMG) ? (2 * IMG - 2 - p) : p;
}
__device__ __forceinline__ unsigned fflip(float f) {  // order-preserving float->uint
  unsigned u = __float_as_uint(f);
  return (u & 0x80000000u) ? ~u : (u | 0x80000000u);
}
__device__ __forceinline__ float funflip(unsigned u) {
  return __uint_as_float((u & 0x80000000u) ? (u ^ 0x80000000u) : ~u);
}

__device__ __forceinline__ v8f wmma_f32_16x16x4(v2f a, v2f b, v8f c) {
  // D = A(16x4,f32) * B(4x16,f32) + C(16x16,f32)   -> v_wmma_f32_16x16x4_f32
  return __builtin_amdgcn_wmma_f32_16x16x4_f32(false, a, false, b, (short)0, c,
                                               false, false);
}

// ---------------------------------------------------------------- init
__global__ void fe_init(unsigned* __restrict__ outU, unsigned* __restrict__ chmax) {
  size_t i = (size_t)blockIdx.x * blockDim.x + threadIdx.x;
  size_t total = (size_t)16 * NPIX;
  for (size_t p = i; p < total; p += (size_t)gridDim.x * blockDim.x) outU[p] = 0u;
  if (i < NCH) chmax[i] = 0u;
}

// ------------------------------------------------- per-octave Gaussian weights
__global__ void __launch_bounds__(256) fe_weights(float* __restrict__ gpAll) {
  int oct = blockIdx.x;
  float sigma = 1.4f * (float)(1 << oct);
  int r = (int)(3.0f * sigma + 0.5f);
  if (r < 1) r = 1;
  int tid = threadIdx.x;
  __shared__ float w[2 * RMAX + 1];
  __shared__ float part[256];
  float local = 0.f;
  for (int i = tid; i < 2 * r + 1; i += 256) {
    float t = (float)(i - r) / sigma;
    float e = expf(-0.5f * t * t);
    w[i] = e;
    local += e;
  }
  part[tid] = local;
  __syncthreads();
  for (int s = 128; s > 0; s >>= 1) {
    if (tid < s) part[tid] += part[tid + s];
    __syncthreads();
  }
  float inv = 1.f / part[0];
  float* gp = gpAll + (size_t)oct * GP_STRIDE;
  for (int i = tid; i < GP_STRIDE; i += 256) gp[i] = 0.f;
  __syncthreads();
  for (int i = tid; i < 2 * r + 1; i += 256) gp[16 + i] = w[i] * inv;
}

// ---------------------------------------------------------- per-channel minmax
__global__ void __launch_bounds__(256) fe_minmax(const float* __restrict__ x,
                                                 float* __restrict__ mnmx) {
  int ch = blockIdx.x;
  const float4* img4 = (const float4*)(x + (size_t)ch * NPIX);
  int tid = threadIdx.x;
  float mn = 3.4e38f, mx = -3.4e38f;
  for (int i = tid; i < NPIX / 4; i += 256) {
    float4 v = img4[i];
    mn = fminf(mn, fminf(fminf(v.x, v.y), fminf(v.z, v.w)));
    mx = fmaxf(mx, fmaxf(fmaxf(v.x, v.y), fmaxf(v.z, v.w)));
  }
  __shared__ float smn[256], smx[256];
  smn[tid] = mn; smx[tid] = mx;
  __syncthreads();
  for (int s = 128; s > 0; s >>= 1) {
    if (tid < s) {
      smn[tid] = fminf(smn[tid], smn[tid + s]);
      smx[tid] = fmaxf(smx[tid], smx[tid + s]);
    }
    __syncthreads();
  }
  if (tid == 0) { mnmx[ch] = smn[0]; mnmx[NCH + ch] = smx[0]; }
}

// --------------------------------- fused: normalize -> Sobel -> Harris R, + Laplacian base
__global__ void __launch_bounds__(256) fe_harris_lap(const float* __restrict__ x,
                                                     const float* __restrict__ mnmx,
                                                     float* __restrict__ R,
                                                     float* __restrict__ lapOut) {
  int ch = blockIdx.z, r0 = blockIdx.y * 16, c0 = blockIdx.x * 16;
  const float* img = x + (size_t)ch * NPIX;
  float mn = mnmx[ch];
  float scale = 255.0f / (mnmx[NCH + ch] - mn + 1e-12f);
  int tid = threadIdx.x;

  __shared__ float g[20][20];                 // gray tile, apron 2, reflect-101
  __shared__ float pxx[18][18], pyy[18][18], pxy[18][18];

  for (int i = tid; i < 400; i += 256) {
    int rr = i / 20, cc = i % 20;
    g[rr][cc] = (img[refl(r0 - 2 + rr) * IMG + refl(c0 - 2 + cc)] - mn) * scale;
  }
  __syncthreads();
  for (int i = tid; i < 324; i += 256) {       // Sobel products at rows r0-1..r0+16
    int rr = i / 18 + 1, cc = i % 18 + 1;
    float ix = (g[rr-1][cc+1] - g[rr-1][cc-1]) + 2.f*(g[rr][cc+1] - g[rr][cc-1])
             + (g[rr+1][cc+1] - g[rr+1][cc-1]);
    float iy = (g[rr+1][cc-1] - g[rr-1][cc-1]) + 2.f*(g[rr+1][cc] - g[rr-1][cc])
             + (g[rr+1][cc+1] - g[rr-1][cc+1]);
    pxx[rr-1][cc-1] = ix * ix;
    pyy[rr-1][cc-1] = iy * iy;
    pxy[rr-1][cc-1] = ix * iy;
  }
  __syncthreads();
  int ty = tid >> 4, tx = tid & 15;
  int iy_ = r0 + ty, ix_ = c0 + tx;
  // 2x2 box over computed products, reflect-101 of the *product arrays* (only -1 -> 1)
  int q0 = iy_ - 1; q0 = (q0 < 0) ? 1 : q0;
  int p0 = ix_ - 1; p0 = (p0 < 0) ? 1 : p0;
  int a0 = q0 - (r0 - 1), a1 = iy_ - (r0 - 1);
  int b0 = p0 - (c0 - 1), b1 = ix_ - (c0 - 1);
  float Sxx = pxx[a0][b0] + pxx[a0][b1] + pxx[a1][b0] + pxx[a1][b1];
  float Syy = pyy[a0][b0] + pyy[a0][b1] + pyy[a1][b0] + pyy[a1][b1];
  float Sxy = pxy[a0][b0] + pxy[a0][b1] + pxy[a1][b0] + pxy[a1][b1];
  float tr = Sxx + Syy;
  R[(size_t)ch * NPIX + iy_ * IMG + ix_] = Sxx * Syy - Sxy * Sxy - HARRIS_K * tr * tr;
  // OpenCV Laplacian ksize=3: 2*(corners) - 8*center
  float lp = 2.f * (g[ty+1][tx+1] + g[ty+1][tx+3] + g[ty+3][tx+1] + g[ty+3][tx+3])
           - 8.f * g[ty+2][tx+2];
  lapOut[(size_t)ch * NPIX + iy_ * IMG + ix_] = lp;
}

// ---------------------------------------------- 3x3 dilate + per-channel max
__global__ void __launch_bounds__(256) fe_dilate(const float* __restrict__ R,
                                                 float* __restrict__ Rd,
                                                 unsigned* __restrict__ chmax) {
  int ch = blockIdx.z, r0 = blockIdx.y * 16, c0 = blockIdx.x * 16;
  const float* Rp = R + (size_t)ch * NPIX;
  __shared__ float t[18][18];
  int tid = threadIdx.x;
  for (int i = tid; i < 324; i += 256) {
    int rr = i / 18, cc = i % 18;
    int gr = r0 - 1 + rr, gc = c0 - 1 + cc;
    t[rr][cc] = (gr >= 0 && gr < IMG && gc >= 0 && gc < IMG) ? Rp[gr * IMG + gc]
                                                             : -3.4e38f;
  }
  __syncthreads();
  int ty = tid >> 4, tx = tid & 15;
  float m = -3.4e38f;
#pragma unroll
  for (int dy = 0; dy < 3; ++dy)
#pragma unroll
    for (int dx = 0; dx < 3; ++dx) m = fmaxf(m, t[ty + dy][tx + dx]);
  Rd[(size_t)ch * NPIX + (r0 + ty) * IMG + (c0 + tx)] = m;
  __shared__ float red[256];
  red[tid] = m;
  __syncthreads();
  for (int s = 128; s > 0; s >>= 1) {
    if (tid < s) red[tid] = fmaxf(red[tid], red[tid + s]);
    __syncthreads();
  }
  if (tid == 0) atomicMax(&chmax[ch], fflip(red[0]));
}

// --------------- WMMA horizontal Gaussian: O(16x16) = X(16xK) * Toeplitz(Kx16), *sigma
// Block: 128 thr = 4 waves, covers 16 rows x 128 cols.
// Each wave computes TWO adjacent 16x16 tiles sharing the Toeplitz (B) fetch:
// 3 LDS dwords per WMMA instead of 4, and 2 independent WMMAs per K-step.
__global__ void __launch_bounds__(128) fe_grow(const float* __restrict__ lap,
                                               float* __restrict__ tmp,
                                               const float* __restrict__ gpAll,
                                               int gpOff, int r, float sigma) {
  __shared__ float slab[16][SLABW2];
  __shared__ float gp[GP_STRIDE];
  int tid = threadIdx.x, lane = tid & 31, wave = tid >> 5;
  int c0 = blockIdx.x * 128, r0 = blockIdx.y * 16, ch = blockIdx.z;
  const float* img = lap + (size_t)ch * NPIX;

  for (int i = tid; i < GP_STRIDE; i += 128) gp[i] = gpAll[gpOff + i];
  const int W = 128 + 2 * r + 4;                      // K padding cols get zero weight
  for (int row = 0; row < 16; ++row)
    for (int col = tid; col < W; col += 128)
      slab[row][col] = img[(r0 + row) * IMG + refl(c0 - r + col)];
  __syncthreads();

  // A 16x4: lanes 0-15 M=lane, K pair {0,1}; lanes 16-31 M=lane-16, K pair {2,3}
  // B 4x16 (rows striped across lanes, C/D-style): VGPR j -> row k0+j (lo half),
  // k0+j+2 (hi half), col = lane&15
  int m = lane & 15, n = lane & 15;
  int kh = (lane >> 4) << 1;                          // 0 or 2
  int bc = wave * 32;                                 // two tiles: bc, bc+16
  v8f acc0 = {0.f, 0.f, 0.f, 0.f, 0.f, 0.f, 0.f, 0.f};
  v8f acc1 = {0.f, 0.f, 0.f, 0.f, 0.f, 0.f, 0.f, 0.f};
  int steps = (2 * r + 19) >> 2;                      // ceil((2r+16)/4)
  for (int s = 0; s < steps; ++s) {
    int k0 = s << 2;
    v2f a0, a1, b;
    b.x = gp[k0 + kh + 0 - n + 16];                   // G[k][n] = g[k-n], 0 outside band
    b.y = gp[k0 + kh + 1 - n + 16];
    a0.x = slab[m][bc + k0 + kh];
    a0.y = slab[m][bc + k0 + kh + 1];
    a1.x = slab[m][bc + 16 + k0 + kh];
    a1.y = slab[m][bc + 16 + k0 + kh + 1];
    acc0 = wmma_f32_16x16x4(a0, b, acc0);
    acc1 = wmma_f32_16x16x4(a1, b, acc1);
  }
  int mbase = (lane >> 4) << 3;                       // C/D: VGPR v -> M=v / v+8
  float* outp = tmp + (size_t)ch * NPIX;
#pragma unroll
  for (int v = 0; v < 8; ++v) {
    outp[(r0 + mbase + v) * IMG + (c0 + bc + n)] = acc0[v] * sigma;
    outp[(r0 + mbase + v) * IMG + (c0 + bc + 16 + n)] = acc1[v] * sigma;
  }
}

// --------- WMMA vertical Gaussian + fused mask*|.| + atomic max into d_out
// Block: 128 thr = 4 waves, covers 128 rows x 16 cols.
// Each wave computes TWO stacked 16x16 tiles sharing the Toeplitz (A) fetch.
__global__ void __launch_bounds__(128) fe_gcol(const float* __restrict__ tmp,
                                               const float* __restrict__ Rd,
                                               const unsigned* __restrict__ chmax,
                                               unsigned* __restrict__ outU,
                                               const float* __restrict__ gpAll,
                                               int gpOff, int r) {
  __shared__ float slab[SLABW2][16];
  __shared__ float gp[GP_STRIDE];
  int tid = threadIdx.x, lane = tid & 31, wave = tid >> 5;
  int c0 = blockIdx.x * 16, r0 = blockIdx.y * 128, ch = blockIdx.z;
  const float* img = tmp + (size_t)ch * NPIX;

  for (int i = tid; i < GP_STRIDE; i += 128) gp[i] = gpAll[gpOff + i];
  const int H = 128 + 2 * r + 4;
  int lc = tid & 15;
  for (int row = tid >> 4; row < H; row += 8)
    slab[row][lc] = img[refl(r0 - r + row) * IMG + c0 + lc];
  __syncthreads();

  int m = lane & 15, n = lane & 15;
  int kh = (lane >> 4) << 1;
  int br = wave * 32;                                 // two tiles: rows br, br+16
  v8f acc0 = {0.f, 0.f, 0.f, 0.f, 0.f, 0.f, 0.f, 0.f};
  v8f acc1 = {0.f, 0.f, 0.f, 0.f, 0.f, 0.f, 0.f, 0.f};
  int steps = (2 * r + 19) >> 2;
  for (int s = 0; s < steps; ++s) {
    int k0 = s << 2;
    v2f a, b0, b1;
    a.x = gp[k0 + kh + 0 - m + 16];                   // A = Toeplitz: G[m][k] = g[k-m]
    a.y = gp[k0 + kh + 1 - m + 16];
    b0.x = slab[br + k0 + kh][n];                     // B = image rows
    b0.y = slab[br + k0 + kh + 1][n];
    b1.x = slab[br + 16 + k0 + kh][n];
    b1.y = slab[br + 16 + k0 + kh + 1][n];
    acc0 = wmma_f32_16x16x4(a, b0, acc0);
    acc1 = wmma_f32_16x16x4(a, b1, acc1);
  }
  float thr = THRESH * funflip(chmax[ch]);
  const float* Rdp = Rd + (size_t)ch * NPIX;
  unsigned* outp = outU + (size_t)(ch / 3) * NPIX;
  int mbase = (lane >> 4) << 3;
#pragma unroll
  for (int v = 0; v < 8; ++v) {
    int y0 = r0 + br + mbase + v, y1 = y0 + 16, xx = c0 + n;
    float z0 = fabsf(acc0[v]);
    float z1 = fabsf(acc1[v]);
    if (Rdp[y0 * IMG + xx] > thr)                     // mask; out>=0 so uint max == float max
      atomicMax(&outp[y0 * IMG + xx], __float_as_uint(z0));
    if (Rdp[y1 * IMG + xx] > thr)
      atomicMax(&outp[y1 * IMG + xx], __float_as_uint(z1));
  }
}

// ---------------------------------------------------------------- host launch
extern "C" void kernel_launch(void* const* d_in, const int* in_sizes, int n_in,
                              void* d_out, int out_size, void* d_ws, size_t ws_size,
                              hipStream_t stream) {
  const float* x = (const float*)d_in[0];
  unsigned* outU = (unsigned*)d_out;
  float* wsf = (float*)d_ws;
  float* mnmx = wsf;                                  // 96 floats
  unsigned* chmax = (unsigned*)(wsf + 96);            // 48 flipped-uint maxima
  float* gpAll = wsf + 160;                           // 6 * GP_STRIDE weights
  float* Rbuf = wsf + 4096;                           // 48*512*512 (reused as tmp)
  float* RdBuf = Rbuf + (size_t)NCH * NPIX;
  float* lapBuf = RdBuf + (size_t)NCH * NPIX;
  float* tmpBuf = Rbuf;                               // R dead after dilate

  fe_init<<<4096, 256, 0, stream>>>(outU, chmax);
  fe_weights<<<6, 256, 0, stream>>>(gpAll);
  fe_minmax<<<NCH, 256, 0, stream>>>(x, mnmx);
  fe_harris_lap<<<dim3(32, 32, NCH), 256, 0, stream>>>(x, mnmx, Rbuf, lapBuf);
  fe_dilate<<<dim3(32, 32, NCH), 256, 0, stream>>>(Rbuf, RdBuf, chmax);
  for (int oct = 0; oct < 6; ++oct) {
    float sigma = 1.4f * (float)(1 << oct);
    int r = (int)(3.0f * sigma + 0.5f);
    if (r < 1) r = 1;
    fe_grow<<<dim3(4, 32, NCH), 128, 0, stream>>>(lapBuf, tmpBuf, gpAll,
                                                  oct * GP_STRIDE, r, sigma);
    fe_gcol<<<dim3(32, 4, NCH), 128, 0, stream>>>(tmpBuf, RdBuf, chmax, outU,
                                                  gpAll, oct * GP_STRIDE, r);
  }
}